// ContrastiveCenterLoss_M_33174327394762
// MI455X (gfx1250) — compile-verified
//
#include <hip/hip_runtime.h>
#include <math.h>

#define DIMH   512
#define NCLS   100
#define POOLG  8
#define GSIZE  64          // 512 / 8 pooled groups
#define NROWP  112         // 100 classes padded to 7 WMMA row-tiles

typedef float v2f __attribute__((ext_vector_type(2)));
typedef float v8f __attribute__((ext_vector_type(8)));

// Workspace layout (floats)
#define WS_YHAT 0          // 100*8 = 800 : normalized pooled centers
#define WS_M    800        // 8          : mean of Yhat rows
#define WS_C0   808        // 64         : constant part of covariance
#define WS_LOSS 1024       // B          : per-batch loss

// ---------------------------------------------------------------------------
// Kernel 1: centers preprocessing (runs once, 1 block).
//   Yhat[c] = normalize(pool8(feature_center[c]))
//   m       = mean_c Yhat[c]
//   C0      = (sum_c Yhat Yhat^T - 100 m m^T) / 199
// ---------------------------------------------------------------------------
__global__ void cc_centers_kernel(const float* __restrict__ fc,
                                  float* __restrict__ ws) {
    __shared__ float sY[NCLS * POOLG];
    __shared__ float sm[POOLG];
    const int t = threadIdx.x;   // 128 threads

    if (t < NCLS) {
        const float4* row = (const float4*)(fc + (size_t)t * DIMH);
        float pool[POOLG];
        float nrm = 0.f;
#pragma unroll
        for (int g = 0; g < POOLG; ++g) {       // static group index
            float sg = 0.f;
#pragma unroll 2
            for (int i = 0; i < GSIZE / 4; ++i) {
                float4 v = row[g * (GSIZE / 4) + i];
                sg += (v.x + v.y) + (v.z + v.w);
            }
            pool[g] = sg * (1.0f / GSIZE);
            nrm += pool[g] * pool[g];
        }
        nrm = sqrtf(nrm) + 1e-6f;
        const float inrm = 1.0f / nrm;
#pragma unroll
        for (int g = 0; g < POOLG; ++g) {
            float v = pool[g] * inrm;
            sY[t * POOLG + g] = v;
            ws[WS_YHAT + t * POOLG + g] = v;
        }
    }
    __syncthreads();
    if (t < POOLG) {
        float s = 0.f;
        for (int c = 0; c < NCLS; ++c) s += sY[c * POOLG + t];
        s *= (1.0f / NCLS);
        sm[t] = s;
        ws[WS_M + t] = s;
    }
    __syncthreads();
    if (t < POOLG * POOLG) {
        const int d = t >> 3, e = t & 7;
        float s = 0.f;
        for (int c = 0; c < NCLS; ++c) s += sY[c * POOLG + d] * sY[c * POOLG + e];
        ws[WS_C0 + t] = (s - (float)NCLS * sm[d] * sm[e]) * (1.0f / (2 * NCLS - 1));
    }
}

// ---------------------------------------------------------------------------
// Kernel 2: one wave (32 lanes) per batch row.
//   - pool + normalize hidden[b] (coalesced float4, shfl combine)
//   - cov = C0 + (50/199) d d^T ; lane-parallel 8x8 Gauss-Jordan inverse
//   - m_dis = diag(D P D^T) entirely on the WMMA pipe:
//       WMMA pair 1: T  = D * P          (zero-padded D[112][8] in LDS)
//       WMMA pair 2: C2 = T * D^T        (A-layout(D) == B-layout(D^T), so
//                                         the a0/a1 registers are reused as B)
//     diag(C2) -> dis; one 32-lane butterfly at the end.
// ---------------------------------------------------------------------------
__global__ void cc_maha_kernel(const float* __restrict__ hidden,
                               const int*   __restrict__ y,
                               const float* __restrict__ wsin,
                               float* __restrict__ loss_out, int B) {
    __shared__ float sY[NCLS * POOLG];       // 800
    __shared__ float sC0[64];
    __shared__ float smv[POOLG];
    __shared__ float sX[8 * POOLG];          // per-wave x-hat
    __shared__ float sP[8 * 64];             // per-wave inverse covariance
    __shared__ float sD[8 * NROWP * POOLG];  // per-wave padded diff matrix
    __shared__ float sT[8 * 16 * 16];        // per-wave T tile (padded cols)

    const int t = threadIdx.x;               // 256 threads = 8 waves
    for (int i = t; i < NCLS * POOLG; i += blockDim.x) sY[i] = wsin[WS_YHAT + i];
    if (t < 64)    sC0[t] = wsin[WS_C0 + t];
    if (t < POOLG) smv[t] = wsin[WS_M + t];
    __syncthreads();

    const int wave = t >> 5, lane = t & 31;
    float* sXw = sX + wave * POOLG;
    float* sPw = sP + wave * 64;
    float* sDw = sD + wave * NROWP * POOLG;
    float* sTw = sT + wave * 256;
    int b = blockIdx.x * 8 + wave;
    if (b >= B) b = B - 1;                   // redundant recompute, deterministic

    // ---- pool hidden[b]: lane covers 16 contiguous floats (4 x float4) ----
    const float4* h4 = (const float4*)(hidden + (size_t)b * DIMH);
    float s = 0.f;
#pragma unroll
    for (int i = 0; i < 4; ++i) {
        float4 v = h4[lane * 4 + i];
        s += (v.x + v.y) + (v.z + v.w);
    }
    s += __shfl_xor(s, 1, 32);
    s += __shfl_xor(s, 2, 32);               // 4-lane subgroup = pooled group

    float smr[POOLG], xh[POOLG], dvr[POOLG];
#pragma unroll
    for (int j = 0; j < POOLG; ++j) smr[j] = smv[j];
    float nrm = 0.f;
#pragma unroll
    for (int g = 0; g < POOLG; ++g) {
        float pg = __shfl(s, g * 4, 32) * (1.0f / GSIZE);
        xh[g] = pg;
        nrm += pg * pg;
    }
    const float inrm = 1.0f / (sqrtf(nrm) + 1e-6f);
#pragma unroll
    for (int g = 0; g < POOLG; ++g) {
        xh[g] *= inrm;
        dvr[g] = xh[g] - smr[g];
    }
    // lane 4g holds pooled group g: static-index store of x-hat to LDS
    if ((lane & 3) == 0) sXw[lane >> 2] = s * (1.0f / GSIZE) * inrm;
    __builtin_amdgcn_wave_barrier();

    // ---- lane-parallel 8x8 in-place Gauss-Jordan inverse ----
    // lane i owns row mi = i&7 (lanes 8..31 compute identical duplicates).
    const int mi = lane & 7;
    const float dvi = sXw[mi] - smv[mi];     // runtime row index via LDS
    float a[8];
#pragma unroll
    for (int j = 0; j < 8; ++j)
        a[j] = sC0[mi * 8 + j] + (50.0f / 199.0f) * dvi * dvr[j];
#pragma unroll
    for (int k = 0; k < 8; ++k) {
        float pr[8];
#pragma unroll
        for (int j = 0; j < 8; ++j) pr[j] = __shfl(a[j], k, 32);
        const float p = 1.0f / pr[k];
        const bool isk = (mi == k);
        const float nc = -a[k] * p;          // new A[i][k], i != k
        a[k] = isk ? a[k] : nc;
#pragma unroll
        for (int j = 0; j < 8; ++j) if (j != k) {
            const float upd = a[j] + a[k] * pr[j];
            a[j] = isk ? a[j] : upd;
        }
#pragma unroll
        for (int j = 0; j < 8; ++j) if (j != k) {
            const float sc = a[j] * p;       // scale pivot row
            a[j] = isk ? sc : a[j];
        }
        a[k] = isk ? p : a[k];
    }
    if (lane < 8) {
#pragma unroll
        for (int j = 0; j < 8; ++j) sPw[mi * 8 + j] = a[j];
    }
    __builtin_amdgcn_wave_barrier();

    // ---- fill zero-padded D[112][8]; lane's column is fixed (= lane&7) ----
    {
        const float xv = sXw[lane & 7];      // 32*i keeps idx&7 constant
        for (int i = 0; i < 25; ++i) {       // 25*32 = 800 valid entries
            const int idx = lane + 32 * i;
            sDw[idx] = xv - sY[idx];
        }
#pragma unroll
        for (int i = 25; i < 28; ++i)        // rows 100..111 -> exact zeros
            sDw[lane + 32 * i] = 0.f;
    }
    __builtin_amdgcn_wave_barrier();

    // ---- WMMA tiles ----
    const int half = lane >> 4;              // C/D row half
    const int col  = lane & 15;              // B/C/D column = lane
    const int c8   = lane & 7;
    const bool colv = (col < 8);
    const int kb = 2 * half;                 // A/B K-pair base per lane half

    // B operand for T = D*P: rows K={kb,kb+1} at column col; P symmetric
    const v2f b0r = *(const v2f*)&sPw[c8 * 8 + kb];
    const v2f b1r = *(const v2f*)&sPw[c8 * 8 + kb + 4];
    v2f b0, b1;
    b0.x = colv ? b0r.x : 0.f;  b0.y = colv ? b0r.y : 0.f;
    b1.x = colv ? b1r.x : 0.f;  b1.y = colv ? b1r.y : 0.f;

    const int yb = y[b];
    const bool diaglane = ((col >> 3) == half);   // lanes holding diag(C2)
    float sumw = 0.f, intraw = 0.f;

#pragma unroll
    for (int tile = 0; tile < 7; ++tile) {
        const int arow = tile * 16 + col;    // A-matrix M = lane&15
        const v2f a0 = *(const v2f*)&sDw[arow * 8 + kb];
        const v2f a1 = *(const v2f*)&sDw[arow * 8 + kb + 4];

        // T = D * P (accumulate over K=8 in two chunks)
        v8f c = {};
        c = __builtin_amdgcn_wmma_f32_16x16x4_f32(false, a0, false, b0,
                                                  (short)0, c, false, false);
        c = __builtin_amdgcn_wmma_f32_16x16x4_f32(false, a1, false, b1,
                                                  (short)0, c, false, false);

        // D-layout -> A-layout via LDS (padded 16x16 so no store guards)
#pragma unroll
        for (int r = 0; r < 8; ++r) sTw[(r + 8 * half) * 16 + col] = c[r];
        __builtin_amdgcn_wave_barrier();

        const v2f t0 = *(const v2f*)&sTw[col * 16 + kb];
        const v2f t1 = *(const v2f*)&sTw[col * 16 + kb + 4];
        __builtin_amdgcn_wave_barrier();

        // C2 = T * D^T : a0/a1 double as the B operand (layout identity)
        v8f c2 = {};
        c2 = __builtin_amdgcn_wmma_f32_16x16x4_f32(false, t0, false, a0,
                                                   (short)0, c2, false, false);
        c2 = __builtin_amdgcn_wmma_f32_16x16x4_f32(false, t1, false, a1,
                                                   (short)0, c2, false, false);

        // diagonal element lives at VGPR r = col&7 on diag lanes
        float md = c2[0];
#pragma unroll
        for (int r = 1; r < 8; ++r) md = (c8 == r) ? c2[r] : md;
        const float dis = sqrtf(fmaxf(md, 0.f));     // 0 for pad rows
        const int ar = tile * 16 + col;
        sumw   += diaglane ? dis : 0.f;
        intraw += (diaglane && ar == yb) ? dis : 0.f;
    }

    // single cross-lane reduction at the end
#pragma unroll
    for (int m = 1; m < 32; m <<= 1) {
        sumw   += __shfl_xor(sumw,   m, 32);
        intraw += __shfl_xor(intraw, m, 32);
    }
    if (lane == 0) {
        loss_out[b] = intraw - (sumw - intraw) * (1.0f / (NCLS - 1));
    }
}

// ---------------------------------------------------------------------------
// Kernel 3: mean over B per-batch losses.
// ---------------------------------------------------------------------------
__global__ void cc_reduce_kernel(const float* __restrict__ loss,
                                 float* __restrict__ out, int B) {
    __shared__ float sh[256];
    const int t = threadIdx.x;
    float s = 0.f;
    for (int i = t; i < B; i += 256) s += loss[i];
    sh[t] = s;
    __syncthreads();
    for (int w = 128; w > 0; w >>= 1) {
        if (t < w) sh[t] += sh[t + w];
        __syncthreads();
    }
    if (t == 0) out[0] = sh[0] / (float)B;
}

extern "C" void kernel_launch(void* const* d_in, const int* in_sizes, int n_in,
                              void* d_out, int out_size, void* d_ws, size_t ws_size,
                              hipStream_t stream) {
    const float* hidden = (const float*)d_in[0];
    const float* fc     = (const float*)d_in[1];
    const int*   y      = (const int*)d_in[2];
    float* out = (float*)d_out;
    float* ws  = (float*)d_ws;
    const int B = in_sizes[0] / DIMH;   // 1024

    cc_centers_kernel<<<1, 128, 0, stream>>>(fc, ws);
    cc_maha_kernel<<<(B + 7) / 8, 256, 0, stream>>>(hidden, y, ws,
                                                    ws + WS_LOSS, B);
    cc_reduce_kernel<<<1, 256, 0, stream>>>(ws + WS_LOSS, out, B);
}